// PointNet2MSG_65412351918081
// MI455X (gfx1250) — compile-verified
//
#include <hip/hip_runtime.h>
#include <hip/hip_bf16.h>

// PointNet++ sample_and_group for MI455X (gfx1250, wave32).
//
// Sizes fixed by setup_inputs(): B=4, N=8192, D=16, S=npoint=2048, K=nsample=32.
//
// d_out layout (reference returns (new_xyz, out), concatenated flat):
//   [0, B*S*3)                      new_xyz  (fp32)
//   [B*S*3, B*S*3 + B*S*K*2D)       out      (fp32)
//
// Kernel 1 (fps_kernel): one 1024-thread block per batch; xyz + running
//   distance arrays LDS-resident (128 KB < 320 KB/WGP); shuffle+LDS argmax,
//   2 barriers per sequential step.
//
// Kernel 2 (knn_group_kernel): one 256-thread block (8 waves) per
//   (batch, 16-query tile).
//   Stage A: distances via V_WMMA_F32_16X16X4_F32 with the norm folded in:
//     A row m = (qx, qy, qz, |q|^2), B col n = (-2px, -2py, -2pz, 1)
//     => D = |q|^2 - 2 q.p ; each lane adds its column's |p|^2.
//     Two 16x16 tiles per wave per chunk (256 candidates/chunk).
//   Stage B: wave-cooperative top-32. Each wave owns 2 query rows; the
//     32-entry pool lives one entry per lane (2 VGPRs/query). 32 candidates
//     are tested per __ballot; inserts replace the pool-max lane and redo a
//     5-step shfl_xor max/argmax. All 8 waves busy in both stages.
//   Then per-query (dist,idx) sort to match lax.top_k ordering, and fused
//   gather/normalize/concat output with all 256 threads.

typedef __attribute__((ext_vector_type(2))) float v2f;
typedef __attribute__((ext_vector_type(8))) float v8f;

namespace {
constexpr int Bb = 4;
constexpr int Nn = 8192;
constexpr int Dd = 16;
constexpr int Ss = 2048;      // npoint
constexpr int Kk = 32;        // nsample
constexpr int QT = 16;        // queries per tile (WMMA M)
constexpr int CHUNK = 256;    // candidates per chunk (8 waves * 2 tiles * 16)
constexpr int CPITCH = 260;   // CHUNK + 4 pad words (bank-conflict-free)
}

// ---------------------------------------------------------------------------
// Kernel 1: farthest point sampling (sequential, LDS-resident)
// ---------------------------------------------------------------------------
__global__ __launch_bounds__(1024) void fps_kernel(const float* __restrict__ xyz,
                                                   float* __restrict__ newXyz,
                                                   int* __restrict__ fpsIdx) {
  __shared__ float xl[Nn];
  __shared__ float yl[Nn];
  __shared__ float zl[Nn];
  __shared__ float dl[Nn];
  __shared__ float wv[32];
  __shared__ int   wi[32];
  __shared__ int   sFar;

  const int b   = blockIdx.x;
  const int tid = threadIdx.x;
  const float* xb = xyz + (size_t)b * Nn * 3;

  for (int i = tid; i < Nn; i += 1024) {
    xl[i] = xb[i * 3 + 0];
    yl[i] = xb[i * 3 + 1];
    zl[i] = xb[i * 3 + 2];
    dl[i] = 1.0e10f;
  }
  if (tid == 0) sFar = 0;
  __syncthreads();

  const int wave = tid >> 5;
  const int lane = tid & 31;

  for (int s = 0; s < Ss; ++s) {
    const int far = sFar;
    if (tid == 0) {
      fpsIdx[b * Ss + s] = far;
      float* nx = newXyz + (size_t)(b * Ss + s) * 3;
      nx[0] = xl[far]; nx[1] = yl[far]; nx[2] = zl[far];
    }
    const float cx = xl[far], cy = yl[far], cz = zl[far];

    float lmax = -1.0f; int lidx = 0;
    for (int i = tid; i < Nn; i += 1024) {
      const float dx = xl[i] - cx, dy = yl[i] - cy, dz = zl[i] - cz;
      float d  = dx * dx + dy * dy + dz * dz;
      float dm = dl[i];
      dm = d < dm ? d : dm;
      dl[i] = dm;
      if (dm > lmax) { lmax = dm; lidx = i; }   // ascending i => first-max kept
    }
    // wave32 argmax (prefer larger value; on tie, smaller index == jnp.argmax)
    #pragma unroll
    for (int off = 16; off > 0; off >>= 1) {
      const float ov = __shfl_xor(lmax, off);
      const int   oi = __shfl_xor(lidx, off);
      if (ov > lmax || (ov == lmax && oi < lidx)) { lmax = ov; lidx = oi; }
    }
    if (lane == 0) { wv[wave] = lmax; wi[wave] = lidx; }
    __syncthreads();
    if (tid == 0) {
      float bm = wv[0]; int bix = wi[0];
      for (int w = 1; w < 32; ++w) {
        const float v = wv[w]; const int ii = wi[w];
        if (v > bm || (v == bm && ii < bix)) { bm = v; bix = ii; }
      }
      sFar = bix;
    }
    __syncthreads();
  }
}

// ---------------------------------------------------------------------------
// Kernel 2: WMMA distance tiles + wave-cooperative top-32 + fused grouping
// ---------------------------------------------------------------------------
__global__ __launch_bounds__(256) void knn_group_kernel(const float* __restrict__ xyz,
                                                        const float* __restrict__ points,
                                                        const int* __restrict__ fpsIdx,
                                                        float* __restrict__ outBuf) {
  __shared__ float qx[QT], qy[QT], qz[QT], q2[QT];
  __shared__ float newPts[QT][Dd];
  __shared__ float distChunk[QT * CPITCH];   // [query][col], 4-word row pad
  __shared__ float knnD[QT][Kk];
  __shared__ int   knnI[QT][Kk];

  const int blk   = blockIdx.x;
  const int b     = blk >> 7;        // 128 tiles per batch (S/QT = 2048/16)
  const int tile  = blk & 127;
  const int sBase = tile * QT;
  const int tid   = threadIdx.x;

  const float* xb = xyz    + (size_t)b * Nn * 3;
  const float* pb = points + (size_t)b * Nn * Dd;

  // --- load query tile (coords + |q|^2) and the query feature rows --------
  if (tid < QT) {
    const int fi = fpsIdx[b * Ss + sBase + tid];
    const float x = xb[fi * 3 + 0], y = xb[fi * 3 + 1], z = xb[fi * 3 + 2];
    qx[tid] = x; qy[tid] = y; qz[tid] = z; q2[tid] = x * x + y * y + z * z;
  }
  {
    const int m = tid >> 4, c = tid & 15;
    const int fi = fpsIdx[b * Ss + sBase + m];
    newPts[m][c] = pb[fi * Dd + c];
  }
  __syncthreads();

  const int lane = tid & 31;
  const int wave = tid >> 5;
  const int mq   = lane & 15;
  const bool hi  = lane >= 16;

  // A operand (16x4 f32, ISA 7.12.2): lanes 0-15 hold K=0,1; lanes 16-31 K=2,3
  v2f aOp;
  aOp.x = hi ? qz[mq] : qx[mq];   // K=2 : K=0
  aOp.y = hi ? q2[mq] : qy[mq];   // K=3 : K=1

  // wave-distributed top-32 pools: this wave owns query rows 2*wave, 2*wave+1;
  // pool entry k lives in lane k.
  float pD[2]; int pI[2]; float curMax[2]; int maxLane[2];
  #pragma unroll
  for (int q = 0; q < 2; ++q) { pD[q] = 3.0e38f; pI[q] = 0; curMax[q] = 3.0e38f; maxLane[q] = 0; }

  for (int cb0 = 0; cb0 < Nn; cb0 += CHUNK) {
    // ---- Stage A: two WMMA tiles per wave -> LDS [query][col] -----------
    #pragma unroll
    for (int t = 0; t < 2; ++t) {
      const int col = 128 * t + wave * QT + mq;
      const int n   = cb0 + col;
      const float px = xb[n * 3 + 0], py = xb[n * 3 + 1], pz = xb[n * 3 + 2];
      const float p2 = px * px + py * py + pz * pz;

      // prefetch next chunk's coords (uniform predicate, no divergence)
      const int pn = (cb0 + CHUNK < Nn) ? (n + CHUNK) : n;
      __builtin_prefetch(xb + pn * 3, 0, 1);

      // B operand (4x16 f32): VGPR0 = K0 / K2 split, VGPR1 = K1 / K3 split
      v2f bOp;
      bOp.x = hi ? (-2.0f * pz) : (-2.0f * px);
      bOp.y = hi ? 1.0f         : (-2.0f * py);

      v8f cz8 = {};
      v8f dacc = __builtin_amdgcn_wmma_f32_16x16x4_f32(
          /*neg_a=*/false, aOp, /*neg_b=*/false, bOp,
          /*c_mod=*/(short)0, cz8, /*reuse_a=*/false, /*reuse_b=*/false);

      const int mrow = hi ? 8 : 0;
      #pragma unroll
      for (int r = 0; r < 8; ++r) {
        distChunk[(mrow + r) * CPITCH + col] = dacc[r] + p2;  // + |p|^2
      }
    }
    __syncthreads();

    // ---- Stage B: wave-cooperative top-32 over this chunk ---------------
    #pragma unroll
    for (int q = 0; q < 2; ++q) {
      const int qr = wave * 2 + q;
      for (int g = 0; g < CHUNK / 32; ++g) {
        const float dval = distChunk[qr * CPITCH + g * 32 + lane];
        unsigned int mask = (unsigned int)__ballot(dval < curMax[q]);
        while (mask) {
          const int l = __builtin_ctz(mask);
          mask &= mask - 1;
          const float d = __shfl(dval, l);       // uniform
          if (d < curMax[q]) {                   // uniform re-check
            const int ci = cb0 + g * 32 + l;
            if (lane == maxLane[q]) { pD[q] = d; pI[q] = ci; }
            // recompute pool max + owning lane (tie: smaller lane)
            float v = pD[q]; int ml = lane;
            #pragma unroll
            for (int off = 16; off > 0; off >>= 1) {
              const float ov = __shfl_xor(v, off);
              const int   ol = __shfl_xor(ml, off);
              if (ov > v || (ov == v && ol < ml)) { v = ov; ml = ol; }
            }
            curMax[q] = v; maxLane[q] = ml;
          }
        }
      }
    }
    __syncthreads();   // before next chunk overwrites distChunk
  }

  // --- dump pools; sort 32 neighbors by (dist, idx) ascending (== top_k) --
  #pragma unroll
  for (int q = 0; q < 2; ++q) {
    const int qr = wave * 2 + q;
    knnD[qr][lane] = pD[q];
    knnI[qr][lane] = pI[q];
  }
  __syncthreads();
  if (tid < QT) {
    for (int k = 0; k < Kk; ++k) {
      int best = k; float bdv = knnD[tid][k]; int biv = knnI[tid][k];
      for (int j = k + 1; j < Kk; ++j) {
        const float dj = knnD[tid][j]; const int ij = knnI[tid][j];
        if (dj < bdv || (dj == bdv && ij < biv)) { best = j; bdv = dj; biv = ij; }
      }
      const float tdk = knnD[tid][k]; const int tik = knnI[tid][k];
      knnD[tid][k] = bdv;  knnI[tid][k] = biv;
      knnD[tid][best] = tdk; knnI[tid][best] = tik;
    }
  }
  __syncthreads();

  // --- fused gather + normalize + concat: out[b,s,k,:] = [p - c, c] -------
  for (int pair = tid; pair < QT * Kk; pair += 256) {
    const int m = pair >> 5, k = pair & 31;
    const int s = sBase + m;
    const int nidx = knnI[m][k];
    const float* pp = pb + (size_t)nidx * Dd;
    float* op = outBuf + ((size_t)((b * Ss + s) * Kk + k)) * (2 * Dd);
    #pragma unroll
    for (int c2 = 0; c2 < Dd; ++c2) {
      const float np = newPts[m][c2];
      op[c2]      = pp[c2] - np;   // grouped_points_norm
      op[Dd + c2] = np;            // center_rep
    }
  }
}

// ---------------------------------------------------------------------------
extern "C" void kernel_launch(void* const* d_in, const int* in_sizes, int n_in,
                              void* d_out, int out_size, void* d_ws, size_t ws_size,
                              hipStream_t stream) {
  const float* xyz    = (const float*)d_in[0];   // [B,N,3]
  const float* points = (const float*)d_in[1];   // [B,N,D]
  // d_in[2]=npoint, d_in[3]=nsample: fixed by setup_inputs, compiled in.

  float* newXyz = (float*)d_out;                       // [B,S,3]
  float* outBuf = (float*)d_out + (size_t)Bb * Ss * 3; // [B,S,K,2D]
  int*   fpsIdx = (int*)d_ws;                          // [B,S] scratch

  fps_kernel<<<Bb, 1024, 0, stream>>>(xyz, newXyz, fpsIdx);
  knn_group_kernel<<<Bb * (Ss / QT), 256, 0, stream>>>(xyz, points, fpsIdx, outBuf);
}